// Model_29987461661224
// MI455X (gfx1250) — compile-verified
//
#include <hip/hip_runtime.h>
#include <math.h>

#define Bsz 16
#define Lsz 2048
#define Hsz 256
#define Nsz 64
#define NLAYERS 6
#define NITERS 4
#define Tc 64
#define Cc 32                      // Lsz / Tc
#define BLH (Bsz*Lsz*Hsz)          // 8388608 elements, layout (H, C, T, B)

typedef float v2f __attribute__((ext_vector_type(2)));
typedef float v8f __attribute__((ext_vector_type(8)));

static __device__ __forceinline__ v8f wmma4(v2f a, v2f b, v8f c) {
  // D = A(16x4) * B(4x16) + C(16x16), fp32 WMMA (CDNA5)
  return __builtin_amdgcn_wmma_f32_16x16x4_f32(false, a, false, b, (short)0, c, false, false);
}

// ---------------------------------------------------------------- utilities
__global__ void zero_kernel(float* __restrict__ p, int n) {
  for (int i = blockIdx.x * blockDim.x + threadIdx.x; i < n; i += gridDim.x * blockDim.x)
    p[i] = 0.0f;
}

__global__ void add_kernel(const float* __restrict__ a, const float* __restrict__ b,
                           float* __restrict__ o) {
  int i = blockIdx.x * 256 + threadIdx.x;
  o[i] = a[i] + b[i];
}

__global__ void relu_copy_kernel(const float* __restrict__ zin, float* __restrict__ z) {
  int i = blockIdx.x * 256 + threadIdx.x;
  z[i] = fmaxf(zin[i], 0.0f);
}

// ------------------------------------------- feed collapse: P/Q (rank-1 GEMM)
__global__ void embfeed_kernel(const float* __restrict__ W_emb, const float* __restrict__ b_emb,
                               const float* __restrict__ W_feed, const float* __restrict__ b_feed,
                               float* __restrict__ PQ) {
  int k = threadIdx.x;                       // 256 threads
  float p = 0.0f, q = 0.0f;
  for (int hh = 0; hh < Hsz; ++hh) {
    float w = W_feed[hh * Hsz + k];
    p += W_emb[hh] * w;
    q += b_emb[hh] * w;
  }
  PQ[k] = p;
  PQ[Hsz + k] = q + b_feed[k];
}

// inj[h][c][s][b] = relu(x[b, c*T+s] * P[h] + Q[h])
__global__ void inj_kernel(const float* __restrict__ x, const float* __restrict__ PQ,
                           float* __restrict__ inj) {
  int idx = blockIdx.x * 256 + threadIdx.x;
  int b = idx & 15;
  int s = (idx >> 4) & 63;
  int c = (idx >> 10) & 31;
  int h = idx >> 15;
  float v = x[b * Lsz + c * Tc + s] * PQ[h] + PQ[Hsz + h];
  inj[idx] = fmaxf(v, 0.0f);
}

// ------------------------------------- per-layer S4D operator precomputation
// E[h][t][n] = C*a^{t+1}; F[h][n][s] = bb*a^{T-1-s}; K[h][d] = sum_n C*bb*a^d; aT = a^T
__global__ void precompute_kernel(const float* __restrict__ log_dt, const float* __restrict__ A_log,
                                  const float* __restrict__ Bm, const float* __restrict__ Cm,
                                  float* __restrict__ E, float* __restrict__ F,
                                  float* __restrict__ K, float* __restrict__ aT, int layer) {
  int h = blockIdx.x;                        // Hsz blocks of Nsz threads
  int n = threadIdx.x;
  __shared__ float tmp[Nsz * 65];
  float dtv = expf(log_dt[layer * Hsz + h]);
  int pidx = (layer * Hsz + h) * Nsz + n;
  float Av = -expf(A_log[pidx]);
  float a = expf(dtv * Av);
  float bb = (a - 1.0f) / Av * Bm[pidx];
  float Cv = Cm[pidx];

  float ap = a;                              // a^{j+1}
  for (int j = 0; j < Tc; ++j) {
    E[((size_t)h * Tc + j) * Nsz + n] = Cv * ap;
    if (j == Tc - 1) aT[h * Nsz + n] = ap;   // a^T
    ap *= a;
  }
  float ap2 = 1.0f;                          // a^d
  for (int d = 0; d < Tc; ++d) {
    F[((size_t)h * Nsz + n) * Tc + (Tc - 1 - d)] = bb * ap2;
    ap2 *= a;
  }
  float p = Cv * bb;
  float ap3 = 1.0f;
  for (int d = 0; d < Tc; ++d) { tmp[n * 65 + d] = p * ap3; ap3 *= a; }
  __syncthreads();
  float s = 0.0f;                            // n reinterpreted as lag d
  for (int m = 0; m < Nsz; ++m) s += tmp[m * 65 + n];
  K[h * Tc + n] = s;
}

// --------------------------- G1: chunk-local end-state sums  S = F_h @ U_chunk
__global__ __launch_bounds__(256) void chunk_state_kernel(const float* __restrict__ U,
                                                          const float* __restrict__ F,
                                                          float* __restrict__ S) {
  int h = blockIdx.x >> 2;
  int ctile = blockIdx.x & 3;
  __shared__ float Fl[Tc * 66];              // padded rows -> no LDS bank conflicts
  for (int i = threadIdx.x; i < Tc * Tc; i += 256) {
    int r = i >> 6, cidx = i & 63;
    Fl[r * 66 + cidx] = F[((size_t)h * Tc + r) * Tc + cidx];
  }
  __syncthreads();
  int w = threadIdx.x >> 5;                  // 8 independent waves, one chunk each
  int lane = threadIdx.x & 31;
  int half = lane >> 4, lq = lane & 15;
  int c = ctile * 8 + w;
  const float* Ub = U + (((size_t)h * Cc + c) << 10);
  float* Sb = S + (((size_t)h * Cc + c) << 10);
  for (int mt = 0; mt < 4; ++mt) {
    v8f acc = {0.f, 0.f, 0.f, 0.f, 0.f, 0.f, 0.f, 0.f};
    int m = mt * 16 + lq;
#pragma unroll
    for (int ks = 0; ks < 16; ++ks) {
      int kb = ks * 4 + half * 2;
      v2f av; av.x = Fl[m * 66 + kb];      av.y = Fl[m * 66 + kb + 1];
      v2f bv; bv.x = Ub[kb * 16 + lq];     bv.y = Ub[(kb + 1) * 16 + lq];
      acc = wmma4(av, bv, acc);
    }
    int rbase = mt * 16 + half * 8;
#pragma unroll
    for (int j = 0; j < 8; ++j) Sb[(rbase + j) * 16 + lq] = acc[j];
  }
}

// -------- G2: sequential prefix over the 32 chunks, in-place S -> x0 per chunk
__global__ void chunk_prefix_kernel(float* __restrict__ S, const float* __restrict__ aT) {
  int t = blockIdx.x * 256 + threadIdx.x;    // H*N*B = 262144 threads
  int b = t & 15;
  int n = (t >> 4) & 63;
  int h = t >> 10;
  float av = aT[h * Nsz + n];
  float x = 0.0f;
  size_t base = (((size_t)h * Cc) << 10) + n * 16 + b;
  for (int c = 0; c < Cc; ++c) {
    float sv = S[base + ((size_t)c << 10)];
    S[base + ((size_t)c << 10)] = x;         // x0 entering chunk c
    x = av * x + sv;
  }
}

// ---- G3: Y = E_h @ x0 + trilToeplitz(K_h) @ U ; + D*u, tanh, BN partial sums
__global__ __launch_bounds__(256) void chunk_output_kernel(const float* __restrict__ U,
                                                           const float* __restrict__ X0,
                                                           const float* __restrict__ E,
                                                           const float* __restrict__ K,
                                                           const float* __restrict__ Dv,
                                                           float* __restrict__ Tout,
                                                           float* __restrict__ bnp) {
  int h = blockIdx.x >> 2;
  int ctile = blockIdx.x & 3;
  __shared__ float El[Tc * 66];
  __shared__ float Kl[Tc];
  __shared__ float r1[256];
  __shared__ float r2[256];
  for (int i = threadIdx.x; i < Tc * Tc; i += 256) {
    int r = i >> 6, cidx = i & 63;
    El[r * 66 + cidx] = E[((size_t)h * Tc + r) * Nsz + cidx];
  }
  if (threadIdx.x < Tc) Kl[threadIdx.x] = K[h * Tc + threadIdx.x];
  __syncthreads();
  int w = threadIdx.x >> 5;
  int lane = threadIdx.x & 31;
  int half = lane >> 4, lq = lane & 15;
  int c = ctile * 8 + w;
  const float* Ub = U + (((size_t)h * Cc + c) << 10);
  const float* Xb = X0 + (((size_t)h * Cc + c) << 10);
  float* Tb = Tout + (((size_t)h * Cc + c) << 10);
  float Dh = Dv[h];
  float s1 = 0.0f, s2 = 0.0f;
  for (int mt = 0; mt < 4; ++mt) {
    v8f acc = {0.f, 0.f, 0.f, 0.f, 0.f, 0.f, 0.f, 0.f};
    int m = mt * 16 + lq;
#pragma unroll
    for (int ks = 0; ks < 16; ++ks) {        // boundary term: E @ x0
      int kb = ks * 4 + half * 2;
      v2f av; av.x = El[m * 66 + kb];      av.y = El[m * 66 + kb + 1];
      v2f bv; bv.x = Xb[kb * 16 + lq];     bv.y = Xb[(kb + 1) * 16 + lq];
      acc = wmma4(av, bv, acc);
    }
#pragma unroll
    for (int ks = 0; ks < 16; ++ks) {        // causal conv: tril-Toeplitz(K) @ U
      int kb = ks * 4 + half * 2;
      int d0 = m - kb;
      v2f av;
      av.x = (d0 >= 0) ? Kl[d0] : 0.0f;      // A-frag synthesized from K, no matrix
      av.y = (d0 >= 1) ? Kl[d0 - 1] : 0.0f;
      v2f bv; bv.x = Ub[kb * 16 + lq];     bv.y = Ub[(kb + 1) * 16 + lq];
      acc = wmma4(av, bv, acc);
    }
    int rbase = mt * 16 + half * 8;
#pragma unroll
    for (int j = 0; j < 8; ++j) {
      int idx = (rbase + j) * 16 + lq;
      float uv = Ub[idx];
      float th = tanhf(acc[j] + Dh * uv);
      Tb[idx] = th;
      s1 += th;
      s2 += th * th;
    }
  }
  r1[threadIdx.x] = s1; r2[threadIdx.x] = s2;
  __syncthreads();
  for (int o = 128; o > 0; o >>= 1) {
    if (threadIdx.x < o) { r1[threadIdx.x] += r1[threadIdx.x + o]; r2[threadIdx.x] += r2[threadIdx.x + o]; }
    __syncthreads();
  }
  if (threadIdx.x == 0) {                    // deterministic per-WG partials (no atomics)
    bnp[(h * 4 + ctile) * 2]     = r1[0];
    bnp[(h * 4 + ctile) * 2 + 1] = r2[0];
  }
}

// ------------------------------------------------------------ BN stats/apply
__global__ void bn_stats_kernel(const float* __restrict__ bnp, const float* __restrict__ gamma_l,
                                float* __restrict__ mu, float* __restrict__ rsg) {
  int h = threadIdx.x;                       // 256 threads
  float s1 = 0.0f, s2 = 0.0f;
  for (int j = 0; j < 4; ++j) { s1 += bnp[(h * 4 + j) * 2]; s2 += bnp[(h * 4 + j) * 2 + 1]; }
  const float inv = 1.0f / (float)(Bsz * Lsz);
  float m = s1 * inv;
  float v = s2 * inv - m * m;                // biased var, matches jnp.var
  mu[h] = m;
  rsg[h] = rsqrtf(v + 1e-5f) * gamma_l[h];
}

__global__ void bn_apply_kernel(float* __restrict__ T, const float* __restrict__ U,
                                const float* __restrict__ mu, const float* __restrict__ rsg,
                                const float* __restrict__ beta_l) {
  int idx = blockIdx.x * 256 + threadIdx.x;
  int h = idx >> 15;
  float t = T[idx];
  T[idx] = (t - mu[h]) * rsg[h] + beta_l[h] + U[idx];   // BN + skip, in-place over t
}

// ----------------------------------------- DEQ residual norms (last iter only)
__global__ void residual_norm_kernel(const float* __restrict__ zin, const float* __restrict__ zold,
                                     float* __restrict__ res2, float* __restrict__ z2) {
  int b = blockIdx.x;                        // one WG per sample -> deterministic
  __shared__ float r1[256];
  __shared__ float r2[256];
  float a1 = 0.0f, a2 = 0.0f;
  for (int k = threadIdx.x; k < BLH / Bsz; k += 256) {
    int idx = k * 16 + b;
    float zn = fmaxf(zin[idx], 0.0f);
    float d = zn - zold[idx];
    a1 += d * d;
    a2 += zn * zn;
  }
  r1[threadIdx.x] = a1; r2[threadIdx.x] = a2;
  __syncthreads();
  for (int o = 128; o > 0; o >>= 1) {
    if (threadIdx.x < o) { r1[threadIdx.x] += r1[threadIdx.x + o]; r2[threadIdx.x] += r2[threadIdx.x + o]; }
    __syncthreads();
  }
  if (threadIdx.x == 0) { res2[b] = r1[0]; z2[b] = r2[0]; }
}

// ------------------------------------------------------------------ epilogue
__global__ void mean_kernel(const float* __restrict__ z, float* __restrict__ meany) {
  int t = blockIdx.x * 256 + threadIdx.x;    // B*H = 4096 threads
  int b = t & 15;
  int h = t >> 4;
  size_t base = ((size_t)h << 15) + b;
  float s = 0.0f;
  for (int ls = 0; ls < Lsz; ++ls) s += z[base + ls * 16];
  meany[h * 16 + b] = s * (1.0f / (float)Lsz);
}

__global__ void head_kernel(const float* __restrict__ meany, const float* __restrict__ W_fc,
                            const float* __restrict__ b_fc, const float* __restrict__ res2,
                            const float* __restrict__ z2, float* __restrict__ out) {
  int t = threadIdx.x;                       // 64 threads
  if (t < 32) {
    int b = t >> 1, k = t & 1;
    float acc = b_fc[k];
    for (int h = 0; h < Hsz; ++h) acc += meany[h * 16 + b] * W_fc[h * 2 + k];
    out[b * 2 + k] = acc;
  } else if (t == 32) {
    float loss = 0.0f, zh = 0.0f;
    for (int b = 0; b < Bsz; ++b) {
      float rel = sqrtf(res2[b]) / (sqrtf(z2[b]) + 1e-8f);
      loss += rel;
      if (rel < 1e-3f) zh += 1.0f;
    }
    out[32] = loss * (1.0f / (float)Bsz);
    out[33] = 4.0f;                          // n_steps = MAX_ITER
    out[34] = zh * (1.0f / (float)Bsz);
  }
}

// ------------------------------------------------------------------- driver
extern "C" void kernel_launch(void* const* d_in, const int* in_sizes, int n_in,
                              void* d_out, int out_size, void* d_ws, size_t ws_size,
                              hipStream_t stream) {
  (void)in_sizes; (void)n_in; (void)out_size; (void)ws_size;
  const float* x      = (const float*)d_in[0];
  const float* W_emb  = (const float*)d_in[1];
  const float* b_emb  = (const float*)d_in[2];
  const float* log_dt = (const float*)d_in[3];
  const float* A_log  = (const float*)d_in[4];
  const float* Bmat   = (const float*)d_in[5];
  const float* Cmat   = (const float*)d_in[6];
  const float* Dvec   = (const float*)d_in[7];
  const float* gamma  = (const float*)d_in[8];
  const float* beta   = (const float*)d_in[9];
  const float* W_feed = (const float*)d_in[10];
  const float* b_feed = (const float*)d_in[11];
  const float* W_fc   = (const float*)d_in[12];
  const float* b_fc   = (const float*)d_in[13];

  float* ws   = (float*)d_ws;                // all buffers in (H, C, T, B) layout
  float* bufA = ws;
  float* bufB = bufA + BLH;
  float* inj  = bufB + BLH;
  float* zbuf = inj + BLH;
  float* Sbuf = zbuf + BLH;                  // H*C*N*B == BLH
  float* Ebuf = Sbuf + BLH;
  float* Fbuf = Ebuf + (size_t)Hsz * Tc * Nsz;
  float* Kbuf = Fbuf + (size_t)Hsz * Nsz * Tc;
  float* aTb  = Kbuf + (size_t)Hsz * Tc;
  float* bnp  = aTb + (size_t)Hsz * Nsz;
  float* mu   = bnp + Hsz * 4 * 2;
  float* rsg  = mu + Hsz;
  float* PQ   = rsg + Hsz;
  float* res2 = PQ + 2 * Hsz;
  float* z2   = res2 + Bsz;
  float* meany = z2 + Bsz;

  const int EW = BLH / 256;                  // elementwise grid

  zero_kernel<<<4096, 256, 0, stream>>>(zbuf, BLH);
  embfeed_kernel<<<1, 256, 0, stream>>>(W_emb, b_emb, W_feed, b_feed, PQ);
  inj_kernel<<<EW, 256, 0, stream>>>(x, PQ, inj);

  for (int it = 0; it < NITERS; ++it) {
    add_kernel<<<EW, 256, 0, stream>>>(zbuf, inj, bufA);   // zin = z + inj
    float* cur = bufA;
    float* oth = bufB;
    for (int l = 0; l < NLAYERS; ++l) {
      precompute_kernel<<<Hsz, Nsz, 0, stream>>>(log_dt, A_log, Bmat, Cmat, Ebuf, Fbuf, Kbuf, aTb, l);
      chunk_state_kernel<<<Hsz * 4, 256, 0, stream>>>(cur, Fbuf, Sbuf);
      chunk_prefix_kernel<<<1024, 256, 0, stream>>>(Sbuf, aTb);
      chunk_output_kernel<<<Hsz * 4, 256, 0, stream>>>(cur, Sbuf, Ebuf, Kbuf, Dvec + l * Hsz, oth, bnp);
      bn_stats_kernel<<<1, 256, 0, stream>>>(bnp, gamma + l * Hsz, mu, rsg);
      bn_apply_kernel<<<EW, 256, 0, stream>>>(oth, cur, mu, rsg, beta + l * Hsz);
      float* tp = cur; cur = oth; oth = tp;
    }
    if (it == NITERS - 1)
      residual_norm_kernel<<<Bsz, 256, 0, stream>>>(cur, zbuf, res2, z2);
    relu_copy_kernel<<<EW, 256, 0, stream>>>(cur, zbuf);   // z = relu(zin)
  }

  mean_kernel<<<16, 256, 0, stream>>>(zbuf, meany);
  head_kernel<<<1, 64, 0, stream>>>(meany, W_fc, b_fc, res2, z2, (float*)d_out);
}